// ParallelLatticeModel_49271864819961
// MI455X (gfx1250) — compile-verified
//
#include <hip/hip_runtime.h>

// CDNA5 / gfx1250: wave32, WMMA f32 16x16x4 for the layer-1 lattice einsum.

typedef __attribute__((ext_vector_type(8))) float v8f;
typedef __attribute__((ext_vector_type(2))) float v2f;

#define D_   16   // input dims
#define L1_  4    // lattices in layer 1
#define E_   24   // ensemble columns
#define K_   16   // calibrator keypoints
#define NV_  16   // 2^4 corners per lattice
#define TR_  16   // batch rows per tile (WMMA M)
#define WAVES_ 8  // waves per block (256 threads)

__device__ __forceinline__ float clamp01(float v) {
    return __builtin_fminf(__builtin_fmaxf(v, 0.0f), 1.0f);
}

__global__ __launch_bounds__(256, 1)
void lattice_ensemble_kernel(const float* __restrict__ x,
                             const float* __restrict__ cal_kp,
                             const float* __restrict__ cal_vals,
                             const float* __restrict__ k1,   // [E][L1][NV]
                             const float* __restrict__ k2,   // [E][1][NV]
                             float* __restrict__ out,        // [B][E]
                             int nrows)
{
    // Calibration tables: xc = v0 + sum_s clamp01(x*a[s] + b[s]) * dv[s]
    __shared__ __align__(16) float sA [D_][K_ - 1];
    __shared__ __align__(16) float sB [D_][K_ - 1];
    __shared__ __align__(16) float sDV[D_][K_ - 1];
    __shared__ float sV0[D_];
    // calibrated-row staging, stride 20 floats: b128-aligned & bank-spread
    __shared__ __align__(16) float sXC[WAVES_][TR_][20];

    const int tid = threadIdx.x;
    if (tid < D_ * (K_ - 1)) {
        int f = tid / (K_ - 1), s = tid % (K_ - 1);
        float kp0 = cal_kp[f * K_ + s], kp1 = cal_kp[f * K_ + s + 1];
        float inv = 1.0f / (kp1 - kp0);
        sA [f][s] = inv;
        sB [f][s] = -kp0 * inv;
        sDV[f][s] = cal_vals[f * K_ + s + 1] - cal_vals[f * K_ + s];
    }
    if (tid < D_) sV0[tid] = cal_vals[tid * K_];
    __syncthreads();

    const int lane = tid & 31;
    const int wid  = tid >> 5;
    const int nn   = lane & 15;           // N index / row index
    const bool hib = (lane >= 16);
    const int  hi  = hib ? 1 : 0;

    // ---- Preload batch-invariant WMMA B fragments (kernels1) ----
    // B 4x16 f32 layout: v0 holds K=0 (lanes0-15) / K=2 (lanes16-31); v1 = K=1/K=3.
    v2f Bf[2][L1_][4];
    #pragma unroll
    for (int ct = 0; ct < 2; ++ct) {
        int e = ct * 16 + nn;
        #pragma unroll
        for (int l = 0; l < L1_; ++l) {
            #pragma unroll
            for (int c = 0; c < 4; ++c) {
                int kb = 4 * c + 2 * hi;
                float b0 = 0.0f, b1 = 0.0f;
                if (e < E_) {
                    const float* kp = k1 + (size_t)e * (L1_ * NV_) + l * NV_ + kb;
                    b0 = kp[0];
                    b1 = kp[1];
                }
                v2f bb; bb[0] = b0; bb[1] = b1;
                Bf[ct][l][c] = bb;
            }
        }
    }
    // kernels2 row for this lane's ensemble column
    float k2r[2][NV_];
    #pragma unroll
    for (int ct = 0; ct < 2; ++ct) {
        int e = ct * 16 + nn;
        #pragma unroll
        for (int k = 0; k < NV_; ++k)
            k2r[ct][k] = (e < E_) ? k2[(size_t)e * NV_ + k] : 0.0f;
    }

    const int ntiles = nrows >> 4;
    const int nwaves = gridDim.x * WAVES_;
    for (int tile = blockIdx.x * WAVES_ + wid; tile < ntiles; tile += nwaves) {
        const int rowBase = tile * TR_;

        // Prefetch next tile's activations into cache hierarchy.
        int nxt = tile + nwaves;
        if (nxt < ntiles)
            __builtin_prefetch(x + (size_t)(nxt * TR_ + nn) * D_ + 8 * hi, 0, 1);

        // ---- 1) Load + calibrate: lane (r=nn, h=hi) does feats 8h..8h+7 of row r
        const float* xp = x + (size_t)(rowBase + nn) * D_ + 8 * hi;
        const float4 xa = *(const float4*)(xp);
        const float4 xb = *(const float4*)(xp + 4);
        float xv[8] = { xa.x, xa.y, xa.z, xa.w, xb.x, xb.y, xb.z, xb.w };
        float xcv[8];
        #pragma unroll
        for (int i = 0; i < 8; ++i) {
            const int f = 8 * hi + i;
            float v = xv[i];
            float acc = sV0[f];
            #pragma unroll
            for (int s = 0; s < K_ - 1; ++s) {
                float t = clamp01(__builtin_fmaf(v, sA[f][s], sB[f][s]));
                acc = __builtin_fmaf(t, sDV[f][s], acc);
            }
            xcv[i] = acc;
        }
        *(float4*)&sXC[wid][nn][8 * hi]     = make_float4(xcv[0], xcv[1], xcv[2], xcv[3]);
        *(float4*)&sXC[wid][nn][8 * hi + 4] = make_float4(xcv[4], xcv[5], xcv[6], xcv[7]);

        // DS ops of a wave complete in order; wait so every lane sees the full row.
        asm volatile("s_wait_dscnt 0" ::: "memory");

        // Each lane re-reads the full calibrated row r=nn (both halves broadcast).
        const float* rp = &sXC[wid][nn][0];
        const float4 r0 = *(const float4*)(rp);
        const float4 r1 = *(const float4*)(rp + 4);
        const float4 r2 = *(const float4*)(rp + 8);
        const float4 r3 = *(const float4*)(rp + 12);
        const float c16[16] = { r0.x, r0.y, r0.z, r0.w,  r1.x, r1.y, r1.z, r1.w,
                                r2.x, r2.y, r2.z, r2.w,  r3.x, r3.y, r3.z, r3.w };

        // ---- 2) Layer-1 multilinear corner weights, dim0 = MSB of corner index
        float w[L1_][NV_];
        #pragma unroll
        for (int l = 0; l < L1_; ++l) {
            const float x0 = c16[4 * l + 0], x1 = c16[4 * l + 1];
            const float x2 = c16[4 * l + 2], x3 = c16[4 * l + 3];
            const float pq[4] = { (1.0f - x0) * (1.0f - x1), (1.0f - x0) * x1,
                                  x0 * (1.0f - x1),          x0 * x1 };
            const float rs[4] = { (1.0f - x2) * (1.0f - x3), (1.0f - x2) * x3,
                                  x2 * (1.0f - x3),          x2 * x3 };
            #pragma unroll
            for (int ii = 0; ii < 4; ++ii)
                #pragma unroll
                for (int jj = 0; jj < 4; ++jj)
                    w[l][ii * 4 + jj] = pq[ii] * rs[jj];
        }

        // ---- WMMA A fragments: 16x4 f32, v0: K=4c (lo) / 4c+2 (hi); v1: +1
        v2f Af[L1_][4];
        #pragma unroll
        for (int l = 0; l < L1_; ++l)
            #pragma unroll
            for (int c = 0; c < 4; ++c) {
                v2f a;
                a[0] = hib ? w[l][4 * c + 2] : w[l][4 * c + 0];
                a[1] = hib ? w[l][4 * c + 3] : w[l][4 * c + 1];
                Af[l][c] = a;
            }

        // ---- 3) h1 GEMM per lattice via chained V_WMMA_F32_16X16X4_F32,
        //         then layer-2 lattice + store, per 16-wide column tile.
        #pragma unroll
        for (int ct = 0; ct < 2; ++ct) {
            v8f acc[L1_];
            #pragma unroll
            for (int l = 0; l < L1_; ++l) acc[l] = (v8f)0.0f;
            #pragma unroll
            for (int l = 0; l < L1_; ++l)
                #pragma unroll
                for (int c = 0; c < 4; ++c)
                    acc[l] = __builtin_amdgcn_wmma_f32_16x16x4_f32(
                        /*neg_a=*/false, Af[l][c],
                        /*neg_b=*/false, Bf[ct][l][c],
                        /*c_mod=*/(short)0, acc[l],
                        /*reuse_a=*/false, /*reuse_b=*/false);

            const int e = ct * 16 + nn;
            #pragma unroll
            for (int j = 0; j < 8; ++j) {
                // lane slot j holds (row = j + 8*hi, col = e): 4 lattice outputs
                const float h0 = clamp01(acc[0][j]);
                const float h1v = clamp01(acc[1][j]);
                const float h2 = clamp01(acc[2][j]);
                const float h3 = clamp01(acc[3][j]);
                const float pq[4] = { (1.0f - h0) * (1.0f - h1v), (1.0f - h0) * h1v,
                                      h0 * (1.0f - h1v),          h0 * h1v };
                const float rs[4] = { (1.0f - h2) * (1.0f - h3), (1.0f - h2) * h3,
                                      h2 * (1.0f - h3),          h2 * h3 };
                float o = 0.0f;
                #pragma unroll
                for (int ii = 0; ii < 4; ++ii)
                    #pragma unroll
                    for (int jj = 0; jj < 4; ++jj)
                        o = __builtin_fmaf(pq[ii] * rs[jj], k2r[ct][ii * 4 + jj], o);
                const int row = j + 8 * hi;
                if (e < E_)
                    out[(size_t)(rowBase + row) * E_ + e] = o;
            }
        }
    }
}

extern "C" void kernel_launch(void* const* d_in, const int* in_sizes, int n_in,
                              void* d_out, int out_size, void* d_ws, size_t ws_size,
                              hipStream_t stream) {
    const float* x        = (const float*)d_in[0];
    const float* cal_kp   = (const float*)d_in[1];
    const float* cal_vals = (const float*)d_in[2];
    const float* kernels1 = (const float*)d_in[3];
    const float* kernels2 = (const float*)d_in[4];
    float* out = (float*)d_out;

    const int nrows  = in_sizes[0] / D_;          // B
    const int ntiles = nrows / TR_;               // 8192 for B=131072
    const int blocks = (ntiles + WAVES_ - 1) / WAVES_;

    lattice_ensemble_kernel<<<blocks, 256, 0, stream>>>(
        x, cal_kp, cal_vals, kernels1, kernels2, out, nrows);
}